// ProjectiveRenderer_2156073583195
// MI455X (gfx1250) — compile-verified
//
#include <hip/hip_runtime.h>

typedef __attribute__((ext_vector_type(2))) float v2f;
typedef __attribute__((ext_vector_type(4))) float v4f;
typedef __attribute__((ext_vector_type(8))) float v8f;

#define ISZ   128
#define HW    (ISZ * ISZ)
#define TSZ   4
#define NEARP 0.1f
#define FARP  100.0f
#define EPSF  1e-9f
#define AMB_I 0.5f
#define DIR_I 0.5f
#define S_LO  (1.0f / FARP)   /* depth FAR  <-> s = 1/FAR  */
#define S_HI  (1.0f / NEARP)  /* depth NEAR <-> s = 1/NEAR */

// ---------------------------------------------------------------------------
// Kernel 1: per-face preprocess.
// Builds 4 rows per face: 3 det-clamped barycentric coef rows and a 4th row
// s_row = sum_k coef[k]/z[k], so one V_WMMA_F32_16X16X4_F32 yields w0..w2 and
// s = sum_k w_k/z_k. Rows are scattered directly into the WMMA lane-packed
// A layout: block q (= f/8) holds 32 lanes x 4 floats; lane L%16 = M row,
// lane/16 selects K pair {0,1}/{2,3}; float pair sub = (f/4)&1 selects which
// of the two 4-face tiles in the b128 the rasterizer loads per lane.
// Faces in [F, Fpad) write zeros -> w=(0,0,0), s=0 -> never valid.
// faceZ record: [z0, z1, z2, light].
// ---------------------------------------------------------------------------
__global__ void pr_prep(const float* __restrict__ T, const float* __restrict__ verts,
                        const int* __restrict__ faces, const float* __restrict__ Km,
                        const float* __restrict__ dist, float* __restrict__ Apk,
                        float* __restrict__ faceZ, int B, int V, int F, int Fpad)
{
    int t = blockIdx.x * blockDim.x + threadIdx.x;
    if (t >= B * Fpad) return;
    int b = t / Fpad, f = t % Fpad;

    float rows[4][4] = {};
    float z0 = 0.f, z1 = 0.f, z2 = 0.f, light = 0.f;

    if (f < F) {
        const float* Tb = T + b * 16;
        const float* Kb = Km + b * 9;
        const float* db = dist + b * 5;
        float dk1 = db[0], dk2 = db[1], dp1 = db[2], dp2 = db[3], dk3 = db[4];

        float u[3], v[3], z[3], wx[3][3];
        for (int k = 0; k < 3; ++k) {
            int vi = faces[((size_t)b * F + f) * 3 + k];
            float X = verts[((size_t)b * V + vi) * 3 + 0];
            float Y = verts[((size_t)b * V + vi) * 3 + 1];
            float Z = verts[((size_t)b * V + vi) * 3 + 2];
            wx[k][0] = X; wx[k][1] = Y; wx[k][2] = Z;
            float cx = Tb[0] * X + Tb[1] * Y + Tb[2]  * Z + Tb[3];
            float cy = Tb[4] * X + Tb[5] * Y + Tb[6]  * Z + Tb[7];
            float cz = Tb[8] * X + Tb[9] * Y + Tb[10] * Z + Tb[11];
            float xp = cx / (cz + EPSF), yp = cy / (cz + EPSF);
            float r2 = xp * xp + yp * yp;
            float rad = 1.f + dk1 * r2 + dk2 * r2 * r2 + dk3 * r2 * r2 * r2;
            float xd = xp * rad + 2.f * dp1 * xp * yp + dp2 * (r2 + 2.f * xp * xp);
            float yd = yp * rad + dp1 * (r2 + 2.f * yp * yp) + 2.f * dp2 * xp * yp;
            float px = Kb[0] * xd + Kb[1] * yd + Kb[2];
            float py = Kb[3] * xd + Kb[4] * yd + Kb[5];
            u[k] = 2.f * (px - ISZ * 0.5f) / ISZ;
            v[k] = 2.f * ((float)ISZ - py - ISZ * 0.5f) / ISZ;
            z[k] = cz;
        }
        // lighting: n = cross(v2-v1, v0-v1), light = AMB + DIR*relu(n.y/|n|)
        float ax = wx[0][0] - wx[1][0], ay = wx[0][1] - wx[1][1], az = wx[0][2] - wx[1][2];
        float bx = wx[2][0] - wx[1][0], by = wx[2][1] - wx[1][1], bz = wx[2][2] - wx[1][2];
        float nx = by * az - bz * ay;
        float ny = bz * ax - bx * az;
        float nz = bx * ay - by * ax;
        float nl = sqrtf(nx * nx + ny * ny + nz * nz) + EPSF;
        float cosv = ny / nl; if (cosv < 0.f) cosv = 0.f;
        light = AMB_I + DIR_I * cosv;

        float x0 = u[0], x1 = u[1], x2 = u[2];
        float y0 = v[0], y1 = v[1], y2 = v[2];
        float det = x0 * (y1 - y2) + x1 * (y2 - y0) + x2 * (y0 - y1);
        if (fabsf(det) < 1e-6f) det = 1e-6f;
        float inv = 1.f / det;
        rows[0][0] = (y1 - y2) * inv; rows[0][1] = (x2 - x1) * inv; rows[0][2] = (x1 * y2 - x2 * y1) * inv;
        rows[1][0] = (y2 - y0) * inv; rows[1][1] = (x0 - x2) * inv; rows[1][2] = (x2 * y0 - x0 * y2) * inv;
        rows[2][0] = (y0 - y1) * inv; rows[2][1] = (x1 - x0) * inv; rows[2][2] = (x0 * y1 - x1 * y0) * inv;
        float iz0 = 1.f / z[0], iz1 = 1.f / z[1], iz2 = 1.f / z[2];
        rows[3][0] = rows[0][0] * iz0 + rows[1][0] * iz1 + rows[2][0] * iz2;
        rows[3][1] = rows[0][1] * iz0 + rows[1][1] * iz1 + rows[2][1] * iz2;
        rows[3][2] = rows[0][2] * iz0 + rows[1][2] * iz1 + rows[2][2] * iz2;
        z0 = z[0]; z1 = z[1]; z2 = z[2];
    }

    // scatter into WMMA lane-packed A layout
    int q = f >> 3, sub = (f >> 2) & 1, l = f & 3;
    float* Ab = Apk + (size_t)b * Fpad * 16 + (size_t)q * 128 + sub * 2;
#pragma unroll
    for (int r = 0; r < 4; ++r) {
        int lane = 4 * l + r;
        v2f lo; lo.x = rows[r][0]; lo.y = rows[r][1];
        v2f hi; hi.x = rows[r][2]; hi.y = rows[r][3];   // rows[r][3] == 0
        *(v2f*)(Ab + (size_t)lane * 4)        = lo;
        *(v2f*)(Ab + (size_t)(lane + 16) * 4) = hi;
    }
    float* Zr = faceZ + ((size_t)b * Fpad + f) * 4;
    Zr[0] = z0; Zr[1] = z1; Zr[2] = z2; Zr[3] = light;
}

// ---------------------------------------------------------------------------
// Branchless per-tile evaluation: one WMMA = 4 faces x 16 pixels; each lane
// owns 2 whole faces. Depth argmin done in s-space (max s) -> no divisions.
// ---------------------------------------------------------------------------
__device__ __forceinline__ void proc_tile(v2f a, v2f bv, int f0, bool hiHalf,
                                          float& bestS, int& bestF,
                                          float& bw0, float& bw1, float& bw2)
{
    v8f cz = {};
    v8f d = __builtin_amdgcn_wmma_f32_16x16x4_f32(
        false, a, false, bv, (short)0, cz, false, false);
    int fbase = f0 + (hiHalf ? 2 : 0);
#pragma unroll
    for (int j = 0; j < 2; ++j) {
        float w0 = d[4 * j + 0], w1 = d[4 * j + 1], w2 = d[4 * j + 2], s = d[4 * j + 3];
        float mn = fminf(fminf(w0, w1), w2);
        float mx = fmaxf(fmaxf(w0, w1), w2);
        bool valid = (mn >= 0.f) & (mx <= 1.f) & (s > S_LO) & (s < S_HI);
        float scand = valid ? s : S_LO;
        bool take = scand > bestS;          // strict > keeps earliest face
        bestS = take ? scand : bestS;
        bestF = take ? (fbase + j) : bestF;
        bw0   = take ? w0 : bw0;
        bw1   = take ? w1 : bw1;
        bw2   = take ? w2 : bw2;
    }
}

// ---------------------------------------------------------------------------
// Kernel 2: rasterize. One wave32 = 16 pixels. Per 16 faces: two coalesced
// b128 loads from the lane-packed table + four WMMAs. No divergent control
// flow anywhere before/inside the WMMA loop (EXEC stays all-ones).
// ---------------------------------------------------------------------------
__global__ void pr_raster(const float* __restrict__ Apk, const float* __restrict__ faceZ,
                          const float* __restrict__ tex, float* __restrict__ out,
                          int B, int F, int Fpad)
{
    const int lane   = threadIdx.x & 31;
    const int wave   = threadIdx.x >> 5;
    const int gw     = blockIdx.x * (blockDim.x >> 5) + wave;
    const int wavesPerBatch = HW / 16;
    const int b      = gw / wavesPerBatch;
    const int pbase  = (gw % wavesPerBatch) * 16;
    const int n      = lane & 15;
    const int p      = pbase + n;
    const int ix     = p & (ISZ - 1);
    const int iy     = p >> 7;
    const float px   = (2.f * ix + 1.f - ISZ) / ISZ;
    const float py   = (2.f * iy + 1.f - ISZ) / ISZ;
    const bool hiHalf = lane >= 16;

    v2f bv;
    bv.x = hiHalf ? 1.f : px;
    bv.y = hiHalf ? 0.f : py;

    const float* Ab = Apk + (size_t)b * Fpad * 16 + (size_t)lane * 4;

    float bestS = S_LO;
    int   bestF = 0;
    float bw0 = 0.f, bw1 = 0.f, bw2 = 0.f;

    for (int fb = 0; fb < Fpad; fb += 16) {
        const float* bp = Ab + ((size_t)(fb >> 3)) * 128;
        v4f A0 = *(const v4f*)(bp);          // faces fb   .. fb+7
        v4f A1 = *(const v4f*)(bp + 128);    // faces fb+8 .. fb+15
        if (fb + 16 < Fpad)                  // uniform branch: prefetch next block
            __builtin_prefetch(bp + 256, 0, 1);
        proc_tile(A0.xy, bv, fb +  0, hiHalf, bestS, bestF, bw0, bw1, bw2);
        proc_tile(A0.zw, bv, fb +  4, hiHalf, bestS, bestF, bw0, bw1, bw2);
        proc_tile(A1.xy, bv, fb +  8, hiHalf, bestS, bestF, bw0, bw1, bw2);
        proc_tile(A1.zw, bv, fb + 12, hiHalf, bestS, bestF, bw0, bw1, bw2);
    }

    // merge the two lane-halves (disjoint face subsets, same pixel)
    float oS = __shfl_xor(bestS, 16, 32);
    int   oF = __shfl_xor(bestF, 16, 32);
    float o0 = __shfl_xor(bw0, 16, 32);
    float o1 = __shfl_xor(bw1, 16, 32);
    float o2 = __shfl_xor(bw2, 16, 32);
    if (oS > bestS || (oS == bestS && oF < bestF)) {
        bestS = oS; bestF = oF; bw0 = o0; bw1 = o1; bw2 = o2;
    }

    if (lane < 16) {
        float cr = 0.f, cg = 0.f, cb = 0.f;
        if (bestS > S_LO) {                  // hit  <=>  depth < FAR
            float zp = 1.f / bestS;          // interpolated depth z = 1/s
            const float* Zr = faceZ + ((size_t)b * Fpad + bestF) * 4;
            float wp0 = bw0 / Zr[0] * zp; wp0 = fminf(fmaxf(wp0, 0.f), 1.f);
            float wp1 = bw1 / Zr[1] * zp; wp1 = fminf(fmaxf(wp1, 0.f), 1.f);
            float wp2 = bw2 / Zr[2] * zp; wp2 = fminf(fmaxf(wp2, 0.f), 1.f);
            float wsum = wp0 + wp1 + wp2 + EPSF;
            wp0 /= wsum; wp1 /= wsum; wp2 /= wsum;
            int t0 = (int)floorf(wp0 * TSZ); t0 = t0 < 0 ? 0 : (t0 > TSZ - 1 ? TSZ - 1 : t0);
            int t1 = (int)floorf(wp1 * TSZ); t1 = t1 < 0 ? 0 : (t1 > TSZ - 1 ? TSZ - 1 : t1);
            int t2 = (int)floorf(wp2 * TSZ); t2 = t2 < 0 ? 0 : (t2 > TSZ - 1 ? TSZ - 1 : t2);
            const float* tp = tex + (((((size_t)b * F + bestF) * TSZ + t0) * TSZ + t1) * TSZ + t2) * 3;
            float light = Zr[3];
            cr = tp[0] * light; cg = tp[1] * light; cb = tp[2] * light;
        }
        size_t ob = (size_t)b * 3 * HW;
        out[ob + 0 * (size_t)HW + p] = cr;
        out[ob + 1 * (size_t)HW + p] = cg;
        out[ob + 2 * (size_t)HW + p] = cb;
    }
}

extern "C" void kernel_launch(void* const* d_in, const int* in_sizes, int n_in,
                              void* d_out, int out_size, void* d_ws, size_t ws_size,
                              hipStream_t stream)
{
    const float* T     = (const float*)d_in[0];
    const float* verts = (const float*)d_in[1];
    const int*   faces = (const int*)d_in[2];
    const float* tex   = (const float*)d_in[3];
    const float* Km    = (const float*)d_in[4];
    const float* dist  = (const float*)d_in[5];
    float*       out   = (float*)d_out;

    const int B    = in_sizes[0] / 16;        // T (B,4,4)
    const int V    = in_sizes[1] / (3 * B);   // vertices (B,V,3)
    const int F    = in_sizes[2] / (3 * B);   // faces (B,F,3)
    const int Fpad = (F + 15) & ~15;          // zero-padded to WMMA granularity

    float* Apk   = (float*)d_ws;                        // B*Fpad*16 floats
    float* faceZ = Apk + (size_t)B * Fpad * 16;         // B*Fpad*4  floats

    const int nprep = B * Fpad;
    pr_prep<<<(nprep + 255) / 256, 256, 0, stream>>>(T, verts, faces, Km, dist,
                                                     Apk, faceZ, B, V, F, Fpad);

    const int nwaves = B * HW / 16;           // one wave per 16 pixels
    pr_raster<<<nwaves / 8, 256, 0, stream>>>(Apk, faceZ, tex, out, B, F, Fpad);
}